// GCN_69475390980196
// MI455X (gfx1250) — compile-verified
//
#include <hip/hip_runtime.h>
#include <cmath>

typedef float v2f __attribute__((ext_vector_type(2)));
typedef float v8f __attribute__((ext_vector_type(8)));

#define D1 128
#define D2 64

// ---------------- init: deg=1 (self loop), agg buffers = 0 ----------------
__global__ void init_ws_kernel(float* __restrict__ dinv, float* __restrict__ agg1,
                               float* __restrict__ agg2, int N) {
  size_t total = (size_t)N * D1;
  size_t stride = (size_t)gridDim.x * blockDim.x;
  for (size_t i = (size_t)blockIdx.x * blockDim.x + threadIdx.x; i < total; i += stride) {
    agg1[i] = 0.0f;
    if (i < (size_t)N * D2) agg2[i] = 0.0f;
    if (i < (size_t)N) dinv[i] = 1.0f;  // self-loop contributes 1 to degree
  }
}

// ---------------- degree over targets ----------------
__global__ void degree_kernel(const int* __restrict__ ei, float* __restrict__ deg, int E) {
  int e = blockIdx.x * blockDim.x + threadIdx.x;
  if (e < E) unsafeAtomicAdd(&deg[ei[E + e]], 1.0f);  // ei[1][e] = target
}

__global__ void rsqrt_kernel(float* __restrict__ deg, int N) {
  int i = blockIdx.x * blockDim.x + threadIdx.x;
  if (i < N) deg[i] = rsqrtf(deg[i]);  // deg >= 1 always (self loops)
}

// ------- fp32 WMMA GEMM with fused row pre-scale: C[m][n] = dscale[m] * (A@B)[m][n] -------
// One wave per 16x16 output tile; K-loop in steps of 4 using v_wmma_f32_16x16x4_f32.
// Requires M % 16 == 0, Ncol % 16 == 0, K % 4 == 0 (holds: 100000/128/64).
__global__ void gemm_f32_wmma_kernel(const float* __restrict__ A, const float* __restrict__ B,
                                     const float* __restrict__ dscale, float* __restrict__ C,
                                     int M, int K, int Ncol) {
  int waveId = threadIdx.x >> 5;
  int lane   = threadIdx.x & 31;
  int tilesN = Ncol >> 4;
  int gw = blockIdx.x * (blockDim.x >> 5) + waveId;
  int rowTile = gw / tilesN;
  int colTile = gw % tilesN;
  if (rowTile * 16 >= M) return;  // wave-uniform: EXEC stays all-ones inside

  int ln   = lane & 15;        // A row / B col within tile
  int half = lane >> 4;        // 0: K = k0,k0+1   1: K = k0+2,k0+3
  int koff = half * 2;

  const float* arow = A + (size_t)(rowTile * 16 + ln) * K;
  const float* bcol = B + (colTile * 16 + ln);

  v8f acc = {};
  for (int k0 = 0; k0 < K; k0 += 4) {
    v2f a, b;
    a.x = arow[k0 + koff];
    a.y = arow[k0 + koff + 1];
    b.x = bcol[(size_t)(k0 + koff) * Ncol];
    b.y = bcol[(size_t)(k0 + koff + 1) * Ncol];
    acc = __builtin_amdgcn_wmma_f32_16x16x4_f32(
        /*neg_a=*/false, a, /*neg_b=*/false, b,
        /*c_mod=*/(short)0, acc, /*reuse_a=*/false, /*reuse_b=*/false);
  }

  // D layout: VGPR v holds rows (v) [lanes 0-15] and (v+8) [lanes 16-31], col = lane&15
  int n = colTile * 16 + ln;
#pragma unroll
  for (int v = 0; v < 8; ++v) {
    int m = rowTile * 16 + v + 8 * half;
    C[(size_t)m * Ncol + n] = acc[v] * dscale[m];
  }
}

// ----- edge scatter, D=128: one wave per edge, float4 per lane, features pre-scaled -----
__global__ void scatter128_kernel(const int* __restrict__ ei, const float* __restrict__ h,
                                  float* __restrict__ agg, int E) {
  int wave = (blockIdx.x * blockDim.x + threadIdx.x) >> 5;
  int lane = threadIdx.x & 31;
  if (wave >= E) return;
  int r = ei[wave];       // source (message)
  int c = ei[E + wave];   // target (aggregate)
  const float4 v = *(const float4*)(h + (size_t)r * D1 + lane * 4);
  float* dst = agg + (size_t)c * D1 + lane * 4;
  unsafeAtomicAdd(dst + 0, v.x);
  unsafeAtomicAdd(dst + 1, v.y);
  unsafeAtomicAdd(dst + 2, v.z);
  unsafeAtomicAdd(dst + 3, v.w);
}

// ----- edge scatter, D=64: one wave per edge, float2 per lane, features pre-scaled -----
__global__ void scatter64_kernel(const int* __restrict__ ei, const float* __restrict__ h,
                                 float* __restrict__ agg, int E) {
  int wave = (blockIdx.x * blockDim.x + threadIdx.x) >> 5;
  int lane = threadIdx.x & 31;
  if (wave >= E) return;
  int r = ei[wave];
  int c = ei[E + wave];
  const float2 v = *(const float2*)(h + (size_t)r * D2 + lane * 2);
  float* dst = agg + (size_t)c * D2 + lane * 2;
  unsafeAtomicAdd(dst + 0, v.x);
  unsafeAtomicAdd(dst + 1, v.y);
}

// --- layer-1 epilogue: self-loop + target scale + bias + relu (in place into agg1) ---
// y1 = relu( dinv[i] * (agg1[i] + h1p[i]) + b1 )   with h1p = dinv[row]-pre-scaled GEMM out
__global__ void act1_kernel(const float* __restrict__ h1p, const float* __restrict__ dinv,
                            const float* __restrict__ b1, float* __restrict__ agg1, int N) {
  size_t idx = (size_t)blockIdx.x * blockDim.x + threadIdx.x;
  if (idx >= (size_t)N * D1) return;
  int i = (int)(idx >> 7);
  int k = (int)(idx & (D1 - 1));
  float v = dinv[i] * (agg1[idx] + h1p[idx]) + b1[k];
  agg1[idx] = v > 0.0f ? v : 0.0f;
}

// ---------------- finalize: self-loop + scale + bias, row L2-normalize, sigmoid ----------
__global__ void finalize_kernel(const float* __restrict__ h2p, const float* __restrict__ dinv,
                                const float* __restrict__ b2, const float* __restrict__ agg2,
                                float* __restrict__ out, int N) {
  int wave = (blockIdx.x * blockDim.x + threadIdx.x) >> 5;
  int lane = threadIdx.x & 31;
  if (wave >= N) return;
  float d = dinv[wave];
  size_t base = (size_t)wave * D2;
  float v0 = d * (agg2[base + lane]      + h2p[base + lane])      + b2[lane];
  float v1 = d * (agg2[base + 32 + lane] + h2p[base + 32 + lane]) + b2[32 + lane];
  float ss = v0 * v0 + v1 * v1;
#pragma unroll
  for (int off = 16; off > 0; off >>= 1) ss += __shfl_xor(ss, off, 32);
  float inv = 1.0f / fmaxf(sqrtf(ss), 1e-12f);
  v0 *= inv;
  v1 *= inv;
  out[base + lane]      = 1.0f / (1.0f + __expf(-v0));
  out[base + 32 + lane] = 1.0f / (1.0f + __expf(-v1));
}

extern "C" void kernel_launch(void* const* d_in, const int* in_sizes, int n_in,
                              void* d_out, int out_size, void* d_ws, size_t ws_size,
                              hipStream_t stream) {
  const float* x  = (const float*)d_in[0];
  const int*   ei = (const int*)d_in[1];   // [2, E] row-major: ei[0..E) = src, ei[E..2E) = dst
  const float* W1 = (const float*)d_in[2];
  const float* b1 = (const float*)d_in[3];
  const float* W2 = (const float*)d_in[4];
  const float* b2 = (const float*)d_in[5];
  int N = in_sizes[0] / D1;
  int E = in_sizes[1] / 2;

  // workspace layout (floats): dinv[N] | h1p[N*128] | agg1[N*128] | h2p[N*64] | agg2[N*64]
  float* ws   = (float*)d_ws;
  float* dinv = ws;
  float* h1p  = dinv + N;
  float* agg1 = h1p + (size_t)N * D1;
  float* h2p  = agg1 + (size_t)N * D1;
  float* agg2 = h2p + (size_t)N * D2;

  init_ws_kernel<<<4096, 256, 0, stream>>>(dinv, agg1, agg2, N);
  degree_kernel<<<(E + 255) / 256, 256, 0, stream>>>(ei, dinv, E);
  rsqrt_kernel<<<(N + 255) / 256, 256, 0, stream>>>(dinv, N);

  // layer 1: h1p = dinv .* (x @ W1)  (WMMA + fused pre-scale), scatter, epilogue
  {
    int waves = (N / 16) * (D1 / 16);
    gemm_f32_wmma_kernel<<<(waves + 7) / 8, 256, 0, stream>>>(x, W1, dinv, h1p, N, D1, D1);
  }
  {
    int blocks = (int)(((size_t)E * 32 + 255) / 256);
    scatter128_kernel<<<blocks, 256, 0, stream>>>(ei, h1p, agg1, E);
  }
  {
    int blocks = (int)(((size_t)N * D1 + 255) / 256);
    act1_kernel<<<blocks, 256, 0, stream>>>(h1p, dinv, b1, agg1, N);
  }

  // layer 2: h2p = dinv .* (y1 @ W2)  (WMMA + fused pre-scale), scatter, finalize
  {
    int waves = (N / 16) * (D2 / 16);
    gemm_f32_wmma_kernel<<<(waves + 7) / 8, 256, 0, stream>>>(agg1, W2, dinv, h2p, N, D1, D2);
  }
  {
    int blocks = (int)(((size_t)E * 32 + 255) / 256);
    scatter64_kernel<<<blocks, 256, 0, stream>>>(ei, h2p, agg2, E);
  }
  {
    int blocks = (N * 32 + 255) / 256;
    finalize_kernel<<<blocks, 256, 0, stream>>>(h2p, dinv, b2, agg2, (float*)d_out, N);
  }
}